// MoEAdapter_56959856279571
// MI455X (gfx1250) — compile-verified
//
#include <hip/hip_runtime.h>
#include <hip/hip_bf16.h>
#include <math.h>

// ---------------------------------------------------------------------------
// MoE adapter, fused for MI455X (gfx1250, wave32).
//  x:[8192,2048] f32, Wr:[2048,8], br:[8], Wd:[8,2048,16], bd:[8,16],
//  Wu:[8,16,2048], bu:[8,2048]  ->  out:[8192,2048] f32
//
// Memory bound (~134MB @ 23.3TB/s ~= 5.8us). One workgroup per 16-token tile:
//   GEMM1: X(16x2048) @ [Wd_all | Wr | 0](2048x144)   v_wmma_f32_16x16x4_f32
//   route: softmax/top-2 via wave32 shfl_xor in 8-lane expert groups
//   GEMM2: G(16x144) @ [Wu_all ; bu ; 0](144x2048)    v_wmma_f32_16x16x4_f32
// All WMMA sits in branch-free code (EXEC all-ones requirement); every wave
// computes the router tile redundantly to keep control flow uniform.
// x is staged to LDS with global_load_async_to_lds_b128 (ASYNCcnt path).
// ---------------------------------------------------------------------------

typedef float v2f __attribute__((ext_vector_type(2)));
typedef float v8f __attribute__((ext_vector_type(8)));
typedef int   v4i __attribute__((ext_vector_type(4)));

#define D_MODEL 2048
#define NEXP    8
#define RANK    16
#define KCH     128          // x LDS chunk (columns)
#define BCOLS   144          // 128 expert cols + 8 router cols + 8 pad
#define KB      144          // phase-B K (128 expert rows + 8 bu rows + 8 pad)

#if defined(__has_builtin)
#if __has_builtin(__builtin_amdgcn_global_load_async_to_lds_b128)
#define USE_ASYNC_LDS 1
#endif
#endif

typedef __attribute__((address_space(1))) v4i* gv4i_p;   // global int4*
typedef __attribute__((address_space(3))) v4i* lv4i_p;   // LDS int4*

__device__ __forceinline__ void wait_async0() {
#if defined(__has_builtin) && __has_builtin(__builtin_amdgcn_s_wait_asynccnt)
    __builtin_amdgcn_s_wait_asynccnt(0);
#else
    asm volatile("s_wait_asynccnt 0" ::: "memory");
#endif
}

// ---- prep: Ball[d][c]: c<128 -> Wd[e=c>>4][d][c&15]; 128..135 -> Wr[d][c-128]; else 0
__global__ void moe_prep_ball(const float* __restrict__ Wd,
                              const float* __restrict__ Wr,
                              float* __restrict__ Ball) {
    int idx = blockIdx.x * blockDim.x + threadIdx.x;
    if (idx >= D_MODEL * BCOLS) return;
    int d = idx / BCOLS, c = idx % BCOLS;
    float v = 0.0f;
    if (c < 128) {
        int e = c >> 4, r = c & 15;
        v = Wd[((size_t)e * D_MODEL + d) * RANK + r];
    } else if (c < 136) {
        v = Wr[(size_t)d * NEXP + (c - 128)];
    }
    Ball[idx] = v;
}

// ---- prep: WuB[r][d]: r<128 -> Wu flat ([E*R, D] row-major); 128..135 -> bu; else 0
__global__ void moe_prep_wub(const float* __restrict__ Wu,
                             const float* __restrict__ bu,
                             float* __restrict__ WuB) {
    int idx = blockIdx.x * blockDim.x + threadIdx.x;
    if (idx >= KB * D_MODEL) return;
    int r = idx / D_MODEL, d = idx % D_MODEL;
    float v = 0.0f;
    if (r < 128)       v = Wu[(size_t)r * D_MODEL + d];
    else if (r < 136)  v = bu[(size_t)(r - 128) * D_MODEL + d];
    WuB[idx] = v;
}

// ---- main fused kernel: 1 block = 16 tokens, 4 waves (128 threads)
__global__ __launch_bounds__(128)
void moe_adapter_fused(const float* __restrict__ x,
                       const float* __restrict__ br,
                       const float* __restrict__ bd,
                       const float* __restrict__ Ball,
                       const float* __restrict__ WuB,
                       float* __restrict__ out) {
    const int lane = threadIdx.x & 31;
    const int wave = threadIdx.x >> 5;
    const int lrow = lane & 15;          // M row (A) or N col (B/C) within half
    const int hi   = (lane >> 4) & 1;    // lanes 16..31: K+2/+3 (A,B), M+8 (C)
    const int tok0 = blockIdx.x * 16;

    __shared__ float xs[16 * KCH];       // staged x chunk (8 KB)
    __shared__ float gs[16 * KB];        // G matrix incl. gate cols (9 KB)

    // ---------------- Phase A: X @ [Wd_all | Wr] ----------------
    // wave w owns expert tiles {w, w+4}; ALL waves compute router tile 8
    // redundantly so control flow stays uniform (WMMA needs EXEC all-ones).
    const int tileE0 = wave;             // expert tile 0..3
    const int tileE1 = wave + 4;         // expert tile 4..7
    v8f acc0 = {}, acc1 = {}, acc2 = {};

    for (int ch = 0; ch < D_MODEL / KCH; ++ch) {
        __syncthreads();   // previous chunk fully consumed
        // coalesced stage: 16 rows x 128 cols = 512 float4, 4 per thread
        #pragma unroll
        for (int j = 0; j < 4; ++j) {
            int fi  = threadIdx.x + j * 128;     // float4 index
            int row = fi >> 5;                   // 32 float4 per row
            int c4  = fi & 31;
            const float* gsrc = x + (size_t)(tok0 + row) * D_MODEL
                                  + ch * KCH + c4 * 4;
            float* ldst = &xs[row * KCH + c4 * 4];
#ifdef USE_ASYNC_LDS
            __builtin_amdgcn_global_load_async_to_lds_b128(
                (gv4i_p)gsrc, (lv4i_p)ldst, 0, 0);
#else
            *(float4*)ldst = *(const float4*)gsrc;
#endif
        }
#ifdef USE_ASYNC_LDS
        wait_async0();
#endif
        __syncthreads();

        for (int kk = 0; kk < KCH; kk += 4) {
            // A fragment (16x4): lane holds K = kk+2*hi, +1 of row lrow
            v2f a = *(const v2f*)&xs[lrow * KCH + kk + 2 * hi];
            int kg = ch * KCH + kk + 2 * hi;
            const float* b0 = Ball + (size_t)kg * BCOLS + lrow;
            const float* b1 = b0 + BCOLS;
            v2f bA, bB, bR;
            bA.x = b0[tileE0 * 16]; bA.y = b1[tileE0 * 16];
            bB.x = b0[tileE1 * 16]; bB.y = b1[tileE1 * 16];
            bR.x = b0[128];         bR.y = b1[128];
            acc0 = __builtin_amdgcn_wmma_f32_16x16x4_f32(
                false, a, false, bA, (short)0, acc0, false, false);
            acc1 = __builtin_amdgcn_wmma_f32_16x16x4_f32(
                false, a, false, bB, (short)0, acc1, false, false);
            acc2 = __builtin_amdgcn_wmma_f32_16x16x4_f32(
                false, a, false, bR, (short)0, acc2, false, false);
        }
    }

    // ---------------- Routing (wave 0 publishes gates; logits in acc2) -----
    if (wave == 0) {
        const float bonus = 3.0f * (1.0f - 1.0f / 5000.0f);  // step 1 of anneal
        const int   e     = lane & 7;                        // expert in 8-lane group
        const float brv   = (lrow < 8) ? br[e] : 0.0f;
        #pragma unroll
        for (int i = 0; i < 8; ++i) {                        // token m = i + hi*8
            float l = acc2[i] + brv + bonus;
            float mx = l;
            mx = fmaxf(mx, __shfl_xor(mx, 1));
            mx = fmaxf(mx, __shfl_xor(mx, 2));
            mx = fmaxf(mx, __shfl_xor(mx, 4));
            float g = __expf(l - mx);
            // top-1 (tie-break: lower index, like lax.top_k)
            float v1 = g; int i1 = e;
            #pragma unroll
            for (int mask = 1; mask <= 4; mask <<= 1) {
                float vo = __shfl_xor(v1, mask);
                int   io = __shfl_xor(i1, mask);
                if (vo > v1 || (vo == v1 && io < i1)) { v1 = vo; i1 = io; }
            }
            // top-2
            float g2 = (e == i1) ? -1.0f : g;   // gates > 0: -1 acts as -inf
            float v2 = g2; int i2 = e;
            #pragma unroll
            for (int mask = 1; mask <= 4; mask <<= 1) {
                float vo = __shfl_xor(v2, mask);
                int   io = __shfl_xor(i2, mask);
                if (vo > v2 || (vo == v2 && io < i2)) { v2 = vo; i2 = io; }
            }
            float denom = v1 + v2;              // softmax denom cancels in renorm
            float wl = (e == i1) ? (v1 / denom)
                     : (e == i2) ? (v2 / denom) : 0.0f;
            if (lrow < 8) {
                int m = i + hi * 8;
                gs[m * KB + 128 + e] = wl;      // gate rows feed bu via GEMM2
                gs[m * KB + 136 + e] = 0.0f;    // K padding
            }
        }
    }
    __syncthreads();

    // ---------------- fold gates: G = w * relu(H + bd) ----------------
    {
        float bd0 = bd[tileE0 * RANK + lrow];
        float bd1 = bd[tileE1 * RANK + lrow];
        #pragma unroll
        for (int i = 0; i < 8; ++i) {
            int m = i + hi * 8;
            float w0 = gs[m * KB + 128 + tileE0];
            float w1 = gs[m * KB + 128 + tileE1];
            gs[m * KB + tileE0 * 16 + lrow] = fmaxf(acc0[i] + bd0, 0.0f) * w0;
            gs[m * KB + tileE1 * 16 + lrow] = fmaxf(acc1[i] + bd1, 0.0f) * w1;
        }
    }
    __syncthreads();

    // ---------------- Phase B: out_tile = G(16x144) @ WuB(144x2048) -------
    // A fragments for all 36 K-steps in registers, reused across 32 N-tiles
    v2f af[KB / 4];
    #pragma unroll
    for (int ks = 0; ks < KB / 4; ++ks)
        af[ks] = *(const v2f*)&gs[lrow * KB + ks * 4 + 2 * hi];

    for (int t = 0; t < 32; ++t) {              // 128 N-tiles over 4 waves
        int ncol = (wave * 32 + t) * 16 + lrow;
        v8f c = {};
        #pragma unroll
        for (int ks = 0; ks < KB / 4; ++ks) {
            int kg = ks * 4 + 2 * hi;
            v2f b;
            b.x = WuB[(size_t)kg * D_MODEL + ncol];
            b.y = WuB[(size_t)(kg + 1) * D_MODEL + ncol];
            c = __builtin_amdgcn_wmma_f32_16x16x4_f32(
                false, af[ks], false, b, (short)0, c, false, false);
        }
        #pragma unroll
        for (int i = 0; i < 8; ++i) {
            int m = i + hi * 8;
            out[(size_t)(tok0 + m) * D_MODEL + ncol] = c[i];
        }
    }
}

extern "C" void kernel_launch(void* const* d_in, const int* in_sizes, int n_in,
                              void* d_out, int out_size, void* d_ws, size_t ws_size,
                              hipStream_t stream) {
    const float* x  = (const float*)d_in[0];
    const float* Wr = (const float*)d_in[1];
    const float* br = (const float*)d_in[2];
    const float* Wd = (const float*)d_in[3];
    const float* bd = (const float*)d_in[4];
    const float* Wu = (const float*)d_in[5];
    const float* bu = (const float*)d_in[6];
    float* out = (float*)d_out;

    float* Ball = (float*)d_ws;                    // [2048 x 144]
    float* WuB  = Ball + (size_t)D_MODEL * BCOLS;  // [144 x 2048]

    {
        int n = D_MODEL * BCOLS;
        moe_prep_ball<<<(n + 255) / 256, 256, 0, stream>>>(Wd, Wr, Ball);
    }
    {
        int n = KB * D_MODEL;
        moe_prep_wub<<<(n + 255) / 256, 256, 0, stream>>>(Wu, bu, WuB);
    }

    int tokens = in_sizes[0] / D_MODEL;            // 8192
    int blocks = tokens / 16;                      // 512 tiles
    moe_adapter_fused<<<blocks, 128, 0, stream>>>(x, br, bd, Ball, WuB, out);
}